// Token_pair_block_81046032875871
// MI455X (gfx1250) — compile-verified
//
#include <hip/hip_runtime.h>
#include <math.h>

#define Bz   16
#define NG   1024
#define NN   1026          // 2 + NG
#define Cc   768
#define Hh   12
#define HD   64
#define CAP  16
#define MTOT (Bz * NN)     // 16416 token rows

typedef float v2f __attribute__((ext_vector_type(2)));
typedef float v8f __attribute__((ext_vector_type(8)));

// ---------------- CDNA5 async global->LDS copy (ASYNCcnt path) ----------------
__device__ __forceinline__ void async_g2l_b128(unsigned lds_off, const void* gptr) {
  // vdst = per-lane LDS byte address, vaddr = 64-bit global address
  asm volatile("global_load_async_to_lds_b128 %0, %1, off"
               :: "v"(lds_off), "v"(gptr)
               : "memory");
}
__device__ __forceinline__ void wait_async0() {
  asm volatile("s_wait_asynccnt 0x0" ::: "memory");
}
__device__ __forceinline__ unsigned lds_off32(const void* p) {
  // generic pointer to LDS: bits [31:0] are the wave-relative LDS offset
  return (unsigned)(size_t)p;
}

// ---------------------------------------------------------------- zero scratch
__global__ void zero_kernel(float* p, int n) {
  int i = blockIdx.x * 256 + threadIdx.x;
  if (i < n) p[i] = 0.0f;
}

// ---------------------------------------------------------------- LayerNorm
__global__ __launch_bounds__(256) void ln_kernel(const float* __restrict__ gen,
                                                 const float* __restrict__ cls,
                                                 const float* __restrict__ box,
                                                 const float* __restrict__ gamma,
                                                 const float* __restrict__ beta,
                                                 float* __restrict__ xln) {
  int row = blockIdx.x;
  int b = row / NN, n = row % NN;
  const float* src = (n == 0) ? (cls + (size_t)b * Cc)
                   : (n == 1) ? (box + (size_t)b * Cc)
                              : (gen + ((size_t)b * NG + (n - 2)) * Cc);
  int t = threadIdx.x;
  float x0 = src[t], x1 = src[t + 256], x2 = src[t + 512];
  __shared__ float red[256];
  red[t] = x0 + x1 + x2;
  __syncthreads();
  for (int o = 128; o > 0; o >>= 1) { if (t < o) red[t] += red[t + o]; __syncthreads(); }
  float mean = red[0] * (1.0f / Cc);
  __syncthreads();
  float d0 = x0 - mean, d1 = x1 - mean, d2 = x2 - mean;
  red[t] = d0 * d0 + d1 * d1 + d2 * d2;
  __syncthreads();
  for (int o = 128; o > 0; o >>= 1) { if (t < o) red[t] += red[t + o]; __syncthreads(); }
  float inv = rsqrtf(red[0] * (1.0f / Cc) + 1e-5f);
  float* dst = xln + (size_t)row * Cc;
  dst[t]       = d0 * inv * gamma[t]       + beta[t];
  dst[t + 256] = d1 * inv * gamma[t + 256] + beta[t + 256];
  dst[t + 512] = d2 * inv * gamma[t + 512] + beta[t + 512];
}

// ---------------------------------------------------------------- K/V GEMM (WMMA + async-LDS pipeline)
// out[m, j] = sum_k xln[m,k] * w_qkv[C + j, k],  j in [0,1536): j<768 -> K, else V.
// Block tile 128(M) x 64(N), Kc = 32, double-buffered LDS filled by
// global_load_async_to_lds_b128; slab k+1 copies overlap slab k WMMAs.
__global__ __launch_bounds__(256) void gemm_kv_kernel(const float* __restrict__ xln,
                                                      const float* __restrict__ wqkv,
                                                      float* __restrict__ Kbuf,
                                                      float* __restrict__ Vbuf) {
  __shared__ float a_lds[2][128][33];   // +1 pad: kill 64-bank conflicts
  __shared__ float b_lds[2][64][33];
  int t = threadIdx.x;
  int lane = t & 31, w = t >> 5;
  int m0 = blockIdx.x * 128;
  int jbase = blockIdx.y * 64;
  const float* wbase = wqkv + (size_t)(Cc + jbase) * Cc;

  v8f acc[4];
  for (int q = 0; q < 4; ++q)
    for (int r = 0; r < 8; ++r) acc[q][r] = 0.0f;

  int mi  = lane & 15;          // M index (A) / N index (B,C) within 16
  int kk2 = (lane >> 4) * 2;    // K pair selected by lane half
  int mr  = w * 16 + mi;

  // Per-thread staging coordinates (16B chunks; contiguous inside a padded row)
  int amm[4], akq[4], bnn[2], bkq[2];
  for (int i = 0; i < 4; ++i) { int f = t + i * 256; amm[i] = f >> 3; akq[i] = (f & 7) * 4; }
  for (int i = 0; i < 2; ++i) { int f = t + i * 256; bnn[i] = f >> 3; bkq[i] = (f & 7) * 4; }

#define STAGE(bufi, k0s)                                                          \
  do {                                                                            \
    for (int i = 0; i < 4; ++i) {                                                 \
      int gm = m0 + amm[i]; if (gm >= MTOT) gm = MTOT - 1;                        \
      async_g2l_b128(lds_off32(&a_lds[bufi][amm[i]][akq[i]]),                     \
                     xln + (size_t)gm * Cc + (k0s) + akq[i]);                     \
    }                                                                             \
    for (int i = 0; i < 2; ++i) {                                                 \
      async_g2l_b128(lds_off32(&b_lds[bufi][bnn[i]][bkq[i]]),                     \
                     wbase + (size_t)bnn[i] * Cc + (k0s) + bkq[i]);               \
    }                                                                             \
  } while (0)

  STAGE(0, 0);                       // prologue fill
  const int iters = Cc / 32;         // 24
  for (int it = 0; it < iters; ++it) {
    wait_async0();                   // own async fills for slab `it` done
    __syncthreads();                 // all waves' fills done, all prior reads done
    if (it + 1 < iters) STAGE((it + 1) & 1, (it + 1) * 32);  // overlap next copy
    int cur = it & 1;
#pragma unroll
    for (int kk = 0; kk < 32; kk += 4) {
      v2f a;
      a.x = a_lds[cur][mr][kk + kk2];
      a.y = a_lds[cur][mr][kk + kk2 + 1];
#pragma unroll
      for (int q = 0; q < 4; ++q) {
        v2f bf;
        bf.x = b_lds[cur][q * 16 + mi][kk + kk2];
        bf.y = b_lds[cur][q * 16 + mi][kk + kk2 + 1];
        acc[q] = __builtin_amdgcn_wmma_f32_16x16x4_f32(
            false, a, false, bf, (short)0, acc[q], false, false);
      }
    }
  }
#undef STAGE

  // Store: C/D layout — acc[q][r] is M = r + 8*(lane>=16), N = lane&15
  int mhalf = (lane >> 4) * 8;
  for (int q = 0; q < 4; ++q) {
    int j = jbase + q * 16 + mi;
    float* outp; int jj;
    if (j < 768) { outp = Kbuf; jj = j; } else { outp = Vbuf; jj = j - 768; }
    int h = jj >> 6, d = jj & 63;
#pragma unroll
    for (int r = 0; r < 8; ++r) {
      int m = m0 + w * 16 + r + mhalf;
      if (m < MTOT) {
        int b = m / NN, n = m % NN;
        outp[(((size_t)b * Hh + h) * NN + n) * HD + d] = acc[q][r];
      }
    }
  }
}

// ---------------------------------------------------------------- rows 0/1 attention
__global__ __launch_bounds__(256) void rows01_kernel(const float* __restrict__ xln,
                                                     const float* __restrict__ wqkv,
                                                     const float* __restrict__ Kbuf,
                                                     const float* __restrict__ Vbuf,
                                                     float* __restrict__ aout,
                                                     float* __restrict__ clss,
                                                     float* __restrict__ boxs) {
  int bh = blockIdx.x; int b = bh / Hh, h = bh % Hh;
  int t = threadIdx.x;
  __shared__ float q_lds[2][HD];
  __shared__ float s_lds[2][NN];
  __shared__ float red[256];

  if (t < 128) {                       // q for rows 0,1 of this head
    int row = t >> 6, d = t & 63;
    const float* xr = xln + (size_t)(b * NN + row) * Cc;
    const float* wr = wqkv + (size_t)(h * HD + d) * Cc;
    float acc = 0.0f;
    for (int k = 0; k < Cc; ++k) acc += xr[k] * wr[k];
    q_lds[row][d] = acc;
  }
  __syncthreads();

  const float* kb = Kbuf + ((size_t)b * Hh + h) * NN * HD;
  for (int n = t; n < NN; n += 256) {
    const float* kr = kb + (size_t)n * HD;
    float s0 = 0.0f, s1 = 0.0f;
    for (int d = 0; d < HD; ++d) { float kv = kr[d]; s0 += q_lds[0][d] * kv; s1 += q_lds[1][d] * kv; }
    s0 *= 0.125f; s1 *= 0.125f;       // hd^-0.5
    if (n == 1) s0 = 0.0f;            // attn[.,.,0,1] = 0.0 logit (pre-softmax)
    if (n == 0) s1 = 0.0f;            // attn[.,.,1,0] = 0.0 logit
    s_lds[0][n] = s0; s_lds[1][n] = s1;
  }
  __syncthreads();

  for (int row = 0; row < 2; ++row) {
    float lm = -3.0e38f;
    for (int n = t; n < NN; n += 256) lm = fmaxf(lm, s_lds[row][n]);
    red[t] = lm; __syncthreads();
    for (int o = 128; o > 0; o >>= 1) { if (t < o) red[t] = fmaxf(red[t], red[t + o]); __syncthreads(); }
    float mx = red[0]; __syncthreads();
    float ls = 0.0f;
    for (int n = t; n < NN; n += 256) { float e = __expf(s_lds[row][n] - mx); s_lds[row][n] = e; ls += e; }
    red[t] = ls; __syncthreads();
    for (int o = 128; o > 0; o >>= 1) { if (t < o) red[t] += red[t + o]; __syncthreads(); }
    float inv = 1.0f / red[0]; __syncthreads();
    for (int n = t; n < NN; n += 256) s_lds[row][n] *= inv;
    __syncthreads();
  }

  const float* vb = Vbuf + ((size_t)b * Hh + h) * NN * HD;
  if (t < 128) {                       // attention output rows 0,1
    int row = t >> 6, d = t & 63;
    float acc = 0.0f;
    for (int n = 0; n < NN; ++n) acc += s_lds[row][n] * vb[(size_t)n * HD + d];
    aout[(size_t)(b * 18 + row) * Cc + h * HD + d] = acc;
  }
  for (int n = 2 + t; n < NN; n += 256) {   // head-summed routing probabilities
    atomicAdd(&clss[b * NG + n - 2], s_lds[0][n]);
    atomicAdd(&boxs[b * NG + n - 2], s_lds[1][n]);
  }
}

// ---------------------------------------------------------------- routing
__global__ __launch_bounds__(512) void route_kernel(const float* __restrict__ clss,
                                                    const float* __restrict__ boxs,
                                                    int* __restrict__ idx) {
  int b = blockIdx.x, t = threadIdx.x;
  __shared__ float key[NG];
  __shared__ int   pid[NG];
  for (int n = t; n < NG; n += 512) { key[n] = -(clss[b * NG + n] + boxs[b * NG + n]); pid[n] = n; }
  __syncthreads();
  for (int ks = 2; ks <= NG; ks <<= 1) {
    for (int jj = ks >> 1; jj > 0; jj >>= 1) {
      for (int i = t; i < NG; i += 512) {
        int ixj = i ^ jj;
        if (ixj > i) {
          bool up = ((i & ks) == 0);
          float ki = key[i], kj = key[ixj];
          bool sw = up ? (ki > kj) : (ki < kj);
          if (sw) { key[i] = kj; key[ixj] = ki; int p = pid[i]; pid[i] = pid[ixj]; pid[ixj] = p; }
        }
      }
      __syncthreads();
    }
  }
  if (t == 0) {
    int boxj[CAP], clsj[CAP];
    for (int s = 0; s < CAP; ++s) { boxj[s] = -1; clsj[s] = -1; }
    int br = 0, cr = 0;
    for (int j = 0; j < NG; ++j) {
      int n = pid[j];
      bool ib = boxs[b * NG + n] > clss[b * NG + n];
      if (ib) { if (br < CAP) boxj[br] = j; br++; }
      else    { if (cr < CAP) clsj[CAP - 1 - cr] = j; cr++; }
    }
    for (int s = 0; s < CAP; ++s) {
      int win = boxj[s] > clsj[s] ? boxj[s] : clsj[s];
      idx[b * CAP + s] = (win >= 0) ? pid[win] : 0;
    }
  }
}

// ---------------------------------------------------------------- Q for selected rows
__global__ __launch_bounds__(256) void qsel_kernel(const float* __restrict__ xln,
                                                   const float* __restrict__ wqkv,
                                                   const int* __restrict__ idx,
                                                   float* __restrict__ qsel) {
  int bs = blockIdx.x; int b = bs >> 4, s = bs & 15;
  int t = threadIdx.x;
  __shared__ float xr[Cc];
  int n = idx[b * CAP + s] + 2;
  const float* src = xln + (size_t)(b * NN + n) * Cc;
  for (int k = t; k < Cc; k += 256) xr[k] = src[k];
  __syncthreads();
  float* dst = qsel + (size_t)bs * Cc;
  for (int c = t; c < Cc; c += 256) {
    const float* wr = wqkv + (size_t)c * Cc;   // Q rows 0..767 of w_qkv
    float acc = 0.0f;
    for (int k = 0; k < Cc; ++k) acc += xr[k] * wr[k];
    dst[c] = acc;
  }
}

// ---------------------------------------------------------------- attention for 16 selected rows
__global__ __launch_bounds__(256) void selattn_kernel(const float* __restrict__ qsel,
                                                      const float* __restrict__ Kbuf,
                                                      const float* __restrict__ Vbuf,
                                                      float* __restrict__ aout) {
  int bh = blockIdx.x; int b = bh / Hh, h = bh % Hh;
  int t = threadIdx.x;
  __shared__ float q_lds[HD];
  __shared__ float s_lds[NN];
  __shared__ float red[256];
  const float* kb = Kbuf + ((size_t)b * Hh + h) * NN * HD;
  const float* vb = Vbuf + ((size_t)b * Hh + h) * NN * HD;
  for (int s = 0; s < CAP; ++s) {
    if (t < HD) q_lds[t] = qsel[(size_t)(b * CAP + s) * Cc + h * HD + t];
    __syncthreads();
    for (int n = t; n < NN; n += 256) {
      const float* kr = kb + (size_t)n * HD;
      float sc = 0.0f;
      for (int d = 0; d < HD; ++d) sc += q_lds[d] * kr[d];
      s_lds[n] = sc * 0.125f;
    }
    __syncthreads();
    float lm = -3.0e38f;
    for (int n = t; n < NN; n += 256) lm = fmaxf(lm, s_lds[n]);
    red[t] = lm; __syncthreads();
    for (int o = 128; o > 0; o >>= 1) { if (t < o) red[t] = fmaxf(red[t], red[t + o]); __syncthreads(); }
    float mx = red[0]; __syncthreads();
    float ls = 0.0f;
    for (int n = t; n < NN; n += 256) { float e = __expf(s_lds[n] - mx); s_lds[n] = e; ls += e; }
    red[t] = ls; __syncthreads();
    for (int o = 128; o > 0; o >>= 1) { if (t < o) red[t] += red[t + o]; __syncthreads(); }
    float inv = 1.0f / red[0]; __syncthreads();
    if (t < HD) {
      float acc = 0.0f;
      for (int n = 0; n < NN; ++n) acc += s_lds[n] * vb[(size_t)n * HD + t];
      aout[(size_t)(b * 18 + 2 + s) * Cc + h * HD + t] = acc * inv;
    }
    __syncthreads();
  }
}

// ---------------------------------------------------------------- projection + residual + scatter
__global__ __launch_bounds__(256) void proj_kernel(const float* __restrict__ aout,
                                                   const float* __restrict__ wproj,
                                                   const float* __restrict__ bproj,
                                                   const float* __restrict__ gen,
                                                   const float* __restrict__ cls,
                                                   const float* __restrict__ box,
                                                   const int* __restrict__ idx,
                                                   float* __restrict__ out) {
  int br = blockIdx.x; int b = br / 18, r = br % 18;
  int t = threadIdx.x;
  __shared__ float a_lds[Cc];
  const float* ar = aout + (size_t)br * Cc;
  for (int k = t; k < Cc; k += 256) a_lds[k] = ar[k];
  __syncthreads();
  const float* catr;
  if (r == 0)      catr = cls + (size_t)b * Cc;
  else if (r == 1) catr = box + (size_t)b * Cc;
  else             catr = gen + ((size_t)b * NG + idx[b * CAP + (r - 2)]) * Cc;
  float* dst;
  if (r == 0)                dst = out + (size_t)(b * 9) * Cc;                     // cls_task row 0
  else if (r >= 2 && r < 10) dst = out + (size_t)(b * 9 + (r - 1)) * Cc;           // cls_task rows 1..8
  else if (r == 1)           dst = out + (size_t)(Bz * 9 + b * 9) * Cc;            // box_task row 0
  else                       dst = out + (size_t)(Bz * 9 + b * 9 + (r - 9)) * Cc;  // box_task rows 1..8
  for (int c = t; c < Cc; c += 256) {
    const float* wr = wproj + (size_t)c * Cc;
    float acc = 0.0f;
    for (int k = 0; k < Cc; ++k) acc += a_lds[k] * wr[k];
    dst[c] = catr[c] + bproj[c] + acc;
  }
}

// ---------------------------------------------------------------- host launcher
extern "C" void kernel_launch(void* const* d_in, const int* in_sizes, int n_in,
                              void* d_out, int out_size, void* d_ws, size_t ws_size,
                              hipStream_t stream) {
  const float* gen   = (const float*)d_in[0];
  const float* cls   = (const float*)d_in[1];
  const float* box   = (const float*)d_in[2];
  const float* wqkv  = (const float*)d_in[3];
  const float* wproj = (const float*)d_in[4];
  const float* bproj = (const float*)d_in[5];
  const float* gamma = (const float*)d_in[6];
  const float* beta  = (const float*)d_in[7];
  float* ws = (float*)d_ws;

  size_t XLN = 0;
  size_t KB  = XLN + (size_t)Bz * NN * Cc;          // layernormed tokens
  size_t VB  = KB  + (size_t)Bz * Hh * NN * HD;     // K
  size_t CL  = VB  + (size_t)Bz * Hh * NN * HD;     // V
  size_t BX  = CL  + (size_t)Bz * NG;               // cls prob sums
  size_t AO  = BX  + (size_t)Bz * NG;               // box prob sums
  size_t QS  = AO  + (size_t)Bz * 18 * Cc;          // attention out (18 rows/batch)
  size_t IX  = QS  + (size_t)Bz * CAP * Cc;         // q of selected rows

  float* xln  = ws + XLN;
  float* Kb   = ws + KB;
  float* Vb   = ws + VB;
  float* clss = ws + CL;
  float* boxs = ws + BX;
  float* aout = ws + AO;
  float* qsel = ws + QS;
  int*   idx  = (int*)(ws + IX);
  float* out  = (float*)d_out;

  zero_kernel<<<(2 * Bz * NG + 255) / 256, 256, 0, stream>>>(clss, 2 * Bz * NG);
  ln_kernel<<<MTOT, 256, 0, stream>>>(gen, cls, box, gamma, beta, xln);
  dim3 g((MTOT + 127) / 128, (2 * Cc) / 64);
  gemm_kv_kernel<<<g, 256, 0, stream>>>(xln, wqkv, Kb, Vb);
  rows01_kernel<<<Bz * Hh, 256, 0, stream>>>(xln, wqkv, Kb, Vb, aout, clss, boxs);
  route_kernel<<<Bz, 512, 0, stream>>>(clss, boxs, idx);
  qsel_kernel<<<Bz * CAP, 256, 0, stream>>>(xln, wqkv, idx, qsel);
  selattn_kernel<<<Bz * Hh, 256, 0, stream>>>(qsel, Kb, Vb, aout);
  proj_kernel<<<Bz * 18, 256, 0, stream>>>(aout, wproj, bproj, gen, cls, box, idx, out);
}